// GraphSAGE_37915971289911
// MI455X (gfx1250) — compile-verified
//
#include <hip/hip_runtime.h>

typedef __attribute__((ext_vector_type(2))) float v2f;
typedef __attribute__((ext_vector_type(8))) float v8f;

#define N_NODES 50000
#define N_EDGES 800000
#define D 64
#define M_TILES (N_NODES / 16)   // 3125, exact

// ---------------------------------------------------------------- zero scratch
__global__ void zero_f32(float* __restrict__ p, long long n) {
    long long i = (long long)blockIdx.x * blockDim.x + threadIdx.x;
    if (i < n) p[i] = 0.0f;
}

// ---------------------------------------------------------------- edge scatter
// 16 threads per edge, 4 floats each -> global_atomic_add_f32 into agg[dst].
__global__ void scatter_accum(const float* __restrict__ x,
                              const int* __restrict__ src,
                              const int* __restrict__ dst,
                              float* __restrict__ agg,
                              float* __restrict__ cnt) {
    long long gid = (long long)blockIdx.x * blockDim.x + threadIdx.x;
    long long e = gid >> 4;
    int j = (int)(gid & 15);
    if (e >= N_EDGES) return;
    int s = src[e];
    int d = dst[e];
    const float4 v = *(const float4*)(x + (long long)s * D + j * 4);
    float* o = agg + (long long)d * D + j * 4;
    atomicAdd(o + 0, v.x);
    atomicAdd(o + 1, v.y);
    atomicAdd(o + 2, v.z);
    atomicAdd(o + 3, v.w);
    if (j == 0) atomicAdd(cnt + d, 1.0f);
}

// ---------------------------------------------------------------- fused SAGE GEMM
// out[16-row tile] = (agg/max(cnt,1)) @ wl + x @ wr + bias, optional ReLU.
// One wave per 16-row M-tile; fp32 WMMA 16x16x4, K=64, N=64 (4 N-tiles).
// Weights are staged once per block into LDS, packed as float2 so each lane's
// B-fragment (K-pair for its column) is one ds_load_b64.
template <bool RELU>
__global__ void __launch_bounds__(256)
sage_gemm(const float* __restrict__ agg,
          const float* __restrict__ cnt,
          const float* __restrict__ xin,
          const float* __restrict__ wl,
          const float* __restrict__ wr,
          const float* __restrict__ bias,
          float* __restrict__ out) {
    // LDS: packed weights. sW[p][n] = { w[2p][n], w[2p+1][n] }, p = 0..31
    __shared__ float2 sWL[32 * D];
    __shared__ float2 sWR[32 * D];
    __shared__ float  sB[D];

    const int tid = threadIdx.x;
#pragma unroll
    for (int i = tid; i < 32 * D; i += 256) {
        int p = i >> 6;            // K-pair index
        int n = i & 63;            // column
        sWL[i] = make_float2(wl[(2 * p) * D + n], wl[(2 * p + 1) * D + n]);
        sWR[i] = make_float2(wr[(2 * p) * D + n], wr[(2 * p + 1) * D + n]);
    }
    if (tid < D) sB[tid] = bias[tid];
    __syncthreads();

    const int wave = blockIdx.x * (blockDim.x >> 5) + (tid >> 5);
    if (wave < M_TILES) {          // wave-uniform: EXEC stays all-1s inside
        const int lane   = tid & 31;
        const int lane16 = lane & 15;
        const int khalf  = lane >> 4;      // 0 or 1

        const int mrow = wave * 16 + lane16;
        const float c   = cnt[mrow];
        const float inv = 1.0f / fmaxf(c, 1.0f);
        const float* aggRow = agg + (long long)mrow * D;
        const float* xRow   = xin + (long long)mrow * D;

        // accumulators for the 4 N-tiles, pre-loaded with bias (N = nt*16+lane16)
        v8f acc[4];
#pragma unroll
        for (int nt = 0; nt < 4; ++nt) {
            float bv = sB[nt * 16 + lane16];
#pragma unroll
            for (int i = 0; i < 8; ++i) acc[nt][i] = bv;
        }

#pragma unroll
        for (int k0 = 0; k0 < D; k0 += 4) {
            const int kb = k0 + khalf * 2;         // this lane's K pair
            const int p  = (k0 >> 1) + khalf;      // packed K-pair index
            v2f aMean, aX;
            aMean.x = aggRow[kb]     * inv;
            aMean.y = aggRow[kb + 1] * inv;
            aX.x = xRow[kb];
            aX.y = xRow[kb + 1];
#pragma unroll
            for (int nt = 0; nt < 4; ++nt) {
                const int ncol = nt * 16 + lane16;
                const float2 fL = sWL[p * D + ncol];   // ds_load_b64
                const float2 fR = sWR[p * D + ncol];   // ds_load_b64
                v2f bL, bR;
                bL.x = fL.x; bL.y = fL.y;
                bR.x = fR.x; bR.y = fR.y;
                acc[nt] = __builtin_amdgcn_wmma_f32_16x16x4_f32(
                    false, aMean, false, bL, (short)0, acc[nt], false, false);
                acc[nt] = __builtin_amdgcn_wmma_f32_16x16x4_f32(
                    false, aX, false, bR, (short)0, acc[nt], false, false);
            }
        }

        // C/D layout: VGPR r -> M = r + 8*khalf, N = nt*16 + lane16
#pragma unroll
        for (int r = 0; r < 8; ++r) {
            const long long row = (long long)wave * 16 + r + khalf * 8;
#pragma unroll
            for (int nt = 0; nt < 4; ++nt) {
                float v = acc[nt][r];
                if (RELU) v = fmaxf(v, 0.0f);
                out[row * D + nt * 16 + lane16] = v;
            }
        }
    }
}

// ---------------------------------------------------------------- launch
extern "C" void kernel_launch(void* const* d_in, const int* in_sizes, int n_in,
                              void* d_out, int out_size, void* d_ws, size_t ws_size,
                              hipStream_t stream) {
    const float* x   = (const float*)d_in[0];
    const int*   ei  = (const int*)d_in[1];     // [2, E]
    const float* w1l = (const float*)d_in[2];
    const float* b1  = (const float*)d_in[3];
    const float* w1r = (const float*)d_in[4];
    const float* w2l = (const float*)d_in[5];
    const float* b2  = (const float*)d_in[6];
    const float* w2r = (const float*)d_in[7];
    float* out = (float*)d_out;

    float* agg = (float*)d_ws;                       // N*D floats
    float* cnt = agg + (size_t)N_NODES * D;          // N floats
    float* h   = cnt + N_NODES;                      // N*D floats

    const int* src = ei;
    const int* dst = ei + N_EDGES;

    const long long zn = (long long)N_NODES * D + N_NODES;
    dim3 zgrid((unsigned)((zn + 255) / 256));
    dim3 sgrid((unsigned)(((long long)N_EDGES * 16 + 255) / 256));
    dim3 ggrid((M_TILES + 7) / 8);                   // 8 waves per 256-thread block

    // ---- layer 1
    zero_f32<<<zgrid, 256, 0, stream>>>(agg, zn);
    scatter_accum<<<sgrid, 256, 0, stream>>>(x, src, dst, agg, cnt);
    sage_gemm<true><<<ggrid, 256, 0, stream>>>(agg, cnt, x, w1l, w1r, b1, h);

    // ---- layer 2
    zero_f32<<<zgrid, 256, 0, stream>>>(agg, zn);
    scatter_accum<<<sgrid, 256, 0, stream>>>(h, src, dst, agg, cnt);
    sage_gemm<false><<<ggrid, 256, 0, stream>>>(agg, cnt, h, w2l, w2r, b2, out);
}